// GCNLayer_84799834292721
// MI455X (gfx1250) — compile-verified
//
#include <hip/hip_runtime.h>
#include <stdint.h>

#define D_FEAT 128
#define NEG_SLOPE 0.5f

constexpr int BLOCK_THREADS   = 256;                                // 8 waves (wave32)
constexpr int WAVES_PER_BLOCK = BLOCK_THREADS / 32;                 // 8
constexpr int EDGES_PER_WAVE  = 32;
constexpr int EDGES_PER_BLOCK = WAVES_PER_BLOCK * EDGES_PER_WAVE;   // 256

// ---- CDNA5 async global->LDS copy (ASYNCcnt-tracked), inline asm so it works
// on both ROCm 7.2 (5-arg builtin) and amdgpu-toolchain (6-arg builtin).
__device__ __forceinline__ void async_g2lds_b128(uint32_t lds_off, const void* gptr) {
    asm volatile("global_load_async_to_lds_b128 %0, %1, off"
                 :: "v"(lds_off), "v"(gptr)
                 : "memory");
}
__device__ __forceinline__ void wait_async0() {
    asm volatile("s_wait_asynccnt 0" ::: "memory");
}
// Low 32 bits of a flat pointer into LDS == the wave-relative LDS byte address
// (ISA 10.2 aperture mapping).
__device__ __forceinline__ uint32_t lds_addr(const void* p) {
    return (uint32_t)(uintptr_t)p;
}

__device__ __forceinline__ void edge_update(const float* __restrict__ embeds,
                                            float* __restrict__ out,
                                            int r, int c, float v, int lane) {
    // One wave gathers exactly one 512B row: 32 lanes x b128, coalesced.
    const float4 x = ((const float4*)(embeds + (size_t)c * D_FEAT))[lane];
    const float4 m = make_float4(v * x.x, v * x.y, v * x.z, v * x.w);
    float* o = out + (size_t)r * D_FEAT + lane * 4;
    __hip_atomic_fetch_add(o + 0, m.x, __ATOMIC_RELAXED, __HIP_MEMORY_SCOPE_AGENT);
    __hip_atomic_fetch_add(o + 1, m.y, __ATOMIC_RELAXED, __HIP_MEMORY_SCOPE_AGENT);
    __hip_atomic_fetch_add(o + 2, m.z, __ATOMIC_RELAXED, __HIP_MEMORY_SCOPE_AGENT);
    __hip_atomic_fetch_add(o + 3, m.w, __ATOMIC_RELAXED, __HIP_MEMORY_SCOPE_AGENT);
}

// First in the file so its assembly shows up in the disasm snippet.
__global__ __launch_bounds__(BLOCK_THREADS)
void scatter_kernel(const int*   __restrict__ rows,
                    const int*   __restrict__ cols,
                    const float* __restrict__ vals,
                    const float* __restrict__ embeds,
                    float*       __restrict__ out,
                    int n_edges) {
    __shared__ __align__(16) int   s_rows[EDGES_PER_BLOCK];
    __shared__ __align__(16) int   s_cols[EDGES_PER_BLOCK];
    __shared__ __align__(16) float s_vals[EDGES_PER_BLOCK];

    const int t         = threadIdx.x;
    const int tileBase  = blockIdx.x * EDGES_PER_BLOCK;
    const int tileEdges = min(EDGES_PER_BLOCK, n_edges - tileBase);
    if (tileEdges <= 0) return;

    const bool fullTile = (tileEdges == EDGES_PER_BLOCK);
    if (fullTile) {
        // Stage the full 256-edge tile via the async global->LDS DMA path:
        // 64 lanes x 16B per array (waves 0-5 each issue one async b128 per lane).
        if (t < 64) {
            async_g2lds_b128(lds_addr(&s_rows[t * 4]), rows + tileBase + t * 4);
        } else if (t < 128) {
            const int u = t - 64;
            async_g2lds_b128(lds_addr(&s_cols[u * 4]), cols + tileBase + u * 4);
        } else if (t < 192) {
            const int u = t - 128;
            async_g2lds_b128(lds_addr(&s_vals[u * 4]), vals + tileBase + u * 4);
        }
        wait_async0();          // per-wave ASYNCcnt -> 0 before the barrier
    } else {
        // Generic tail tile: direct loads.
        if (t < tileEdges) {
            s_rows[t] = rows[tileBase + t];
            s_cols[t] = cols[tileBase + t];
            s_vals[t] = vals[tileBase + t];
        }
    }
    __syncthreads();

    const int wave = t >> 5;
    const int lane = t & 31;
    const int base = wave * EDGES_PER_WAVE;

    if (fullTile) {
        // Hot path: no per-iteration bounds check.
        for (int i = 0; i < EDGES_PER_WAVE; ++i) {
            const int idx = base + i;
            // Prefetch the next edge's embedding row while gathering this one.
            // 8 lanes per 128B line coalesce to one L2 request.
            if (i + 1 < EDGES_PER_WAVE) {
                const int cn = s_cols[idx + 1];
                __builtin_prefetch(embeds + (size_t)cn * D_FEAT + lane * 4, 0, 3);
            }
            edge_update(embeds, out, s_rows[idx], s_cols[idx], s_vals[idx], lane);
        }
    } else {
        for (int i = 0; i < EDGES_PER_WAVE; ++i) {
            const int idx = base + i;
            if (idx >= tileEdges) break;
            edge_update(embeds, out, s_rows[idx], s_cols[idx], s_vals[idx], lane);
        }
    }
}

__global__ __launch_bounds__(BLOCK_THREADS)
void zero_kernel(float4* __restrict__ out4, int n4) {
    int i = blockIdx.x * BLOCK_THREADS + threadIdx.x;
    if (i < n4) out4[i] = make_float4(0.f, 0.f, 0.f, 0.f);
}

__global__ __launch_bounds__(BLOCK_THREADS)
void lrelu_kernel(float4* __restrict__ out4, int n4) {
    int i = blockIdx.x * BLOCK_THREADS + threadIdx.x;
    if (i < n4) {
        float4 a = out4[i];
        a.x = a.x > 0.f ? a.x : NEG_SLOPE * a.x;
        a.y = a.y > 0.f ? a.y : NEG_SLOPE * a.y;
        a.z = a.z > 0.f ? a.z : NEG_SLOPE * a.z;
        a.w = a.w > 0.f ? a.w : NEG_SLOPE * a.w;
        out4[i] = a;
    }
}

extern "C" void kernel_launch(void* const* d_in, const int* in_sizes, int n_in,
                              void* d_out, int out_size, void* d_ws, size_t ws_size,
                              hipStream_t stream) {
    const int*   rows   = (const int*)d_in[0];
    const int*   cols   = (const int*)d_in[1];
    const float* vals   = (const float*)d_in[2];
    const float* embeds = (const float*)d_in[3];
    float*       out    = (float*)d_out;

    const int E  = in_sizes[0];
    const int n4 = out_size / 4;                       // float4 count (N*D/4)
    const int vb = (n4 + BLOCK_THREADS - 1) / BLOCK_THREADS;
    const int sb = (E + EDGES_PER_BLOCK - 1) / EDGES_PER_BLOCK;

    zero_kernel   <<<vb, BLOCK_THREADS, 0, stream>>>((float4*)out, n4);
    scatter_kernel<<<sb, BLOCK_THREADS, 0, stream>>>(rows, cols, vals, embeds, out, E);
    lrelu_kernel  <<<vb, BLOCK_THREADS, 0, stream>>>((float4*)out, n4);
}